// MyInstanceNorm1d_83124797047432
// MI455X (gfx1250) — compile-verified
//
#include <hip/hip_runtime.h>

// Causal unbiased-EMA instance norm, x: [2048 channels][16384 t] f32.
// Memory-bound: 256MiB min traffic -> ~11us @ 23.3TB/s. Implemented as a
// 3-pass chunk-parallel linear-recurrence scan; the 16-step intra-tile scans
// are done with v_wmma_f32_16x16x4_f32 (S = L*X + c), data staged via
// global_load_async_to_lds / global_store_async_from_lds (double buffered).

typedef float v2f __attribute__((ext_vector_type(2)));
typedef float v4f __attribute__((ext_vector_type(4)));
typedef float v8f __attribute__((ext_vector_type(8)));

namespace {
constexpr float kAlpha = 0.99f;
constexpr float kEps   = 1e-5f;
constexpr int kT      = 16384;
constexpr int kNCh    = 2048;          // B*C = 8*256
constexpr int kLc     = 512;           // chunk length along T
constexpr int kNChunk = kT / kLc;      // 32
constexpr int kStg    = 64;            // timesteps staged per LDS buffer
constexpr int kNStg   = kLc / kStg;    // 8
constexpr int kP      = 68;            // LDS row pitch (floats), conflict-free
constexpr int kInFloats = 16 * kP;     // 1088 floats per staged tile
constexpr int kInBytes  = kInFloats * 4;  // 4352 (16B aligned)
static_assert(kInBytes == 4352, "lds tile size");
}

__device__ __forceinline__ void async_load_b128(unsigned lds, unsigned voff,
                                                const void* sbase) {
  asm volatile("global_load_async_to_lds_b128 %0, %1, %2"
               :: "v"(lds), "v"(voff), "s"(sbase) : "memory");
}
__device__ __forceinline__ void async_store_b128(unsigned voff, unsigned lds,
                                                 void* sbase) {
  asm volatile("global_store_async_from_lds_b128 %0, %1, %2"
               :: "v"(voff), "v"(lds), "s"(sbase) : "memory");
}
__device__ __forceinline__ void wait_async0() {
  asm volatile("s_wait_asynccnt 0" ::: "memory");
}
__device__ __forceinline__ void wait_ds0() {
  asm volatile("s_wait_dscnt 0" ::: "memory");
}

// ---------------- Pass 1: per-(chgroup, chunk) local decayed sums ----------
__global__ __launch_bounds__(128) void k_pass1(const float* __restrict__ x,
                                               float* __restrict__ ws1,
                                               float* __restrict__ ws2) {
  __shared__ __align__(16) char smem[4 * 2 * kInBytes];
  const int tid  = threadIdx.x;
  const int wv   = tid >> 5;
  const int lane = tid & 31;
  const int n    = lane & 15;   // channel within group / WMMA M (=A row)
  const int h    = lane >> 4;   // lane half
  const int tile = blockIdx.x * 4 + wv;          // [0, 4096)
  const int chunk  = tile & (kNChunk - 1);
  const int chbase = (tile >> 5) * 16;           // 128 channel groups
  const int t0     = chunk * kLc;

  const unsigned shbase = (unsigned)(size_t)(&smem[0]) + (unsigned)(wv * 2 * kInBytes);
  unsigned voff[8], lds0[8];
#pragma unroll
  for (int i = 0; i < 8; ++i) {
    voff[i] = (unsigned)(((2 * i + h) * kT + n * 4) * 4);
    lds0[i] = shbase + (unsigned)(((2 * i + h) * kP + n * 4) * 4);
  }
  const float* gx = x + (size_t)chbase * kT + t0;

#pragma unroll
  for (int i = 0; i < 8; ++i) async_load_b128(lds0[i], voff[i], gx);  // stage 0

  // Constant A = L (16x16 lower-tri, K-sliced): L[m,k] = (1-a)*a^(m-k), k<=m
  v2f a[4];
#pragma unroll
  for (int j = 0; j < 4; ++j) {
    const int k0 = j * 4 + 2 * h;
    a[j].x = (k0     <= n) ? (1.0f - kAlpha) * __powf(kAlpha, (float)(n - k0))     : 0.0f;
    a[j].y = (k0 + 1 <= n) ? (1.0f - kAlpha) * __powf(kAlpha, (float)(n - k0 - 1)) : 0.0f;
  }
  float ccoef[8];               // a^(t_sub+1) for C/D row t_sub = r + 8h
#pragma unroll
  for (int r = 0; r < 8; ++r) ccoef[r] = __powf(kAlpha, (float)(r + 8 * h + 1));

  float c1 = 0.0f, c2 = 0.0f;   // chunk-local state (zero entry)

#pragma unroll 1
  for (int s = 0; s < kNStg; ++s) {
    wait_async0();                              // stage s resident in LDS
    if (s + 1 < kNStg) {                        // prefetch next stage
      const unsigned bofs = (unsigned)(((s + 1) & 1) * kInBytes);
#pragma unroll
      for (int i = 0; i < 8; ++i)
        async_load_b128(lds0[i] + bofs, voff[i], gx + (size_t)(s + 1) * kStg);
    }
    const float* row = (const float*)smem + wv * 2 * kInFloats +
                       (s & 1) * kInFloats + n * kP;
#pragma unroll
    for (int sub = 0; sub < 4; ++sub) {
      const float* tp = row + sub * 16;
      v8f s1, s2;
#pragma unroll
      for (int r = 0; r < 8; ++r) { s1[r] = ccoef[r] * c1; s2[r] = ccoef[r] * c2; }
#pragma unroll
      for (int j = 0; j < 4; ++j) {
        v2f b = *(const v2f*)(tp + j * 4 + 2 * h);       // ds_load_b64, K-slice
        s1 = __builtin_amdgcn_wmma_f32_16x16x4_f32(false, a[j], false, b,
                                                   (short)0, s1, false, false);
        v2f b2; b2.x = b.x * b.x; b2.y = b.y * b.y;
        s2 = __builtin_amdgcn_wmma_f32_16x16x4_f32(false, a[j], false, b2,
                                                   (short)0, s2, false, false);
      }
      c1 = __shfl(s1[7], 16 + n, 32);   // D row M=15 -> per-channel carry
      c2 = __shfl(s2[7], 16 + n, 32);
    }
  }
  if (h == 0) {
    ws1[chunk * kNCh + chbase + n] = c1;
    ws2[chunk * kNCh + chbase + n] = c2;
  }
}

// ---------------- Pass 2: exclusive cross-chunk carry scan -----------------
__global__ void k_pass2(float* __restrict__ ws1, float* __restrict__ ws2) {
  const int ch = blockIdx.x * blockDim.x + threadIdx.x;
  if (ch >= kNCh) return;
  const float d = __powf(kAlpha, (float)kLc);
  float c1 = 0.0f, c2 = 0.0f;
#pragma unroll 1
  for (int c = 0; c < kNChunk; ++c) {
    const int idx = c * kNCh + ch;
    const float l1 = ws1[idx], l2 = ws2[idx];
    ws1[idx] = c1; ws2[idx] = c2;          // in-place: entry state of chunk c
    c1 = d * c1 + l1;
    c2 = d * c2 + l2;
  }
}

// ---------------- Pass 3: full scan + normalize + output -------------------
__global__ __launch_bounds__(128) void k_pass3(const float* __restrict__ x,
                                               const float* __restrict__ ws1,
                                               const float* __restrict__ ws2,
                                               float* __restrict__ out) {
  __shared__ __align__(16) char smem[4 * 3 * kInBytes];  // in0|in1|out per wave
  const int tid  = threadIdx.x;
  const int wv   = tid >> 5;
  const int lane = tid & 31;
  const int n    = lane & 15;
  const int h    = lane >> 4;
  const int tile = blockIdx.x * 4 + wv;
  const int chunk  = tile & (kNChunk - 1);
  const int chbase = (tile >> 5) * 16;
  const int t0     = chunk * kLc;

  const unsigned shbase = (unsigned)(size_t)(&smem[0]) + (unsigned)(wv * 3 * kInBytes);
  unsigned voff[8], lds0[8], ldso[8];
#pragma unroll
  for (int i = 0; i < 8; ++i) {
    voff[i] = (unsigned)(((2 * i + h) * kT + n * 4) * 4);
    const unsigned r = (unsigned)(((2 * i + h) * kP + n * 4) * 4);
    lds0[i] = shbase + r;
    ldso[i] = shbase + (unsigned)(2 * kInBytes) + r;
  }
  const float* gx  = x   + (size_t)chbase * kT + t0;
  float*       gout = out + (size_t)chbase * kT + t0;

#pragma unroll
  for (int i = 0; i < 8; ++i) async_load_b128(lds0[i], voff[i], gx);

  v2f a[4];
#pragma unroll
  for (int j = 0; j < 4; ++j) {
    const int k0 = j * 4 + 2 * h;
    a[j].x = (k0     <= n) ? (1.0f - kAlpha) * __powf(kAlpha, (float)(n - k0))     : 0.0f;
    a[j].y = (k0 + 1 <= n) ? (1.0f - kAlpha) * __powf(kAlpha, (float)(n - k0 - 1)) : 0.0f;
  }
  float ccoef[8];
#pragma unroll
  for (int r = 0; r < 8; ++r) ccoef[r] = __powf(kAlpha, (float)(r + 8 * h + 1));

  float c1 = ws1[chunk * kNCh + chbase + n];   // carry-in state
  float c2 = ws2[chunk * kNCh + chbase + n];
  float apow = __powf(kAlpha, (float)t0);      // a^(subtile t base); w = 1-a^(t+1)
  const float a16 = __powf(kAlpha, 16.0f);

#pragma unroll 1
  for (int s = 0; s < kNStg; ++s) {
    wait_async0();                    // stage-s input ready; out buf drained
    if (s + 1 < kNStg) {
#pragma unroll
      for (int i = 0; i < 8; ++i)
        async_load_b128(lds0[i] + (unsigned)(((s + 1) & 1) * kInBytes), voff[i],
                        gx + (size_t)(s + 1) * kStg);
    }
    const float* row  = (const float*)smem + wv * 3 * kInFloats +
                        (s & 1) * kInFloats + n * kP;
    float*       orow = (float*)smem + wv * 3 * kInFloats + 2 * kInFloats + n * kP;
#pragma unroll
    for (int sub = 0; sub < 4; ++sub) {
      const float* tp  = row  + sub * 16;
      float*       otp = orow + sub * 16;
      const v4f xlo = *(const v4f*)(tp + 8 * h);        // C-layout x, rows r=0..3
      const v4f xhi = *(const v4f*)(tp + 8 * h + 4);    // rows r=4..7
      v8f s1, s2;
#pragma unroll
      for (int r = 0; r < 8; ++r) { s1[r] = ccoef[r] * c1; s2[r] = ccoef[r] * c2; }
#pragma unroll
      for (int j = 0; j < 4; ++j) {
        v2f b = *(const v2f*)(tp + j * 4 + 2 * h);
        s1 = __builtin_amdgcn_wmma_f32_16x16x4_f32(false, a[j], false, b,
                                                   (short)0, s1, false, false);
        v2f b2; b2.x = b.x * b.x; b2.y = b.y * b.y;
        s2 = __builtin_amdgcn_wmma_f32_16x16x4_f32(false, a[j], false, b2,
                                                   (short)0, s2, false, false);
      }
      c1 = __shfl(s1[7], 16 + n, 32);
      c2 = __shfl(s2[7], 16 + n, 32);
      float o[8];
#pragma unroll
      for (int r = 0; r < 8; ++r) {
        const float w    = 1.0f - apow * ccoef[r];      // 1 - a^(t+1)
        const float rw   = 1.0f / w;
        const float mean = s1[r] * rw;
        const float var  = s2[r] * rw - mean * mean;
        const float xv   = (r < 4) ? xlo[r] : xhi[r - 4];
        o[r] = (xv - mean) * rsqrtf(var + kEps);
      }
      apow *= a16;
      const v4f olo = {o[0], o[1], o[2], o[3]};
      const v4f ohi = {o[4], o[5], o[6], o[7]};
      *(v4f*)(otp + 8 * h)     = olo;                   // ds_store_b128
      *(v4f*)(otp + 8 * h + 4) = ohi;
    }
    wait_ds0();                        // out tile committed to LDS
#pragma unroll
    for (int i = 0; i < 8; ++i)
      async_store_b128(voff[i], ldso[i], gout + (size_t)s * kStg);
  }
  // s_endpgm performs implicit wait-idle for outstanding async stores
}

extern "C" void kernel_launch(void* const* d_in, const int* in_sizes, int n_in,
                              void* d_out, int out_size, void* d_ws, size_t ws_size,
                              hipStream_t stream) {
  (void)in_sizes; (void)n_in; (void)out_size; (void)ws_size;
  const float* x = (const float*)d_in[0];
  float* out = (float*)d_out;
  float* ws1 = (float*)d_ws;                 // [32][2048] chunk states (s1)
  float* ws2 = ws1 + kNChunk * kNCh;         // [32][2048] chunk states (s2); 512KB total
  k_pass1<<<1024, 128, 0, stream>>>(x, ws1, ws2);
  k_pass2<<<kNCh / 256, 256, 0, stream>>>(ws1, ws2);
  k_pass3<<<1024, 128, 0, stream>>>(x, ws1, ws2, out);
}